// GATWithFeatureAttentionVT_9045201125950
// MI455X (gfx1250) — compile-verified
//
#include <hip/hip_runtime.h>
#include <hip/hip_bf16.h>

typedef __attribute__((ext_vector_type(2))) float v2f;
typedef __attribute__((ext_vector_type(8))) float v8f;

#define GAT_IN 128
#define GAT_H  4
#define GAT_C  16
#define GAT_HC 64
#define NEG_SLOPE 0.2f

// ---------------------------------------------------------------------------
// Init: out = bias (broadcast), copy edge_index bits into d_out, maxv = -inf,
// denom = 0. One grid covering the largest region.
// ---------------------------------------------------------------------------
__global__ void gat_init(float* __restrict__ out, int* __restrict__ ei_out,
                         const int* __restrict__ ei_in,
                         const float* __restrict__ bias,
                         float* __restrict__ maxv, float* __restrict__ denom,
                         int n, int e2) {
  int i = blockIdx.x * blockDim.x + threadIdx.x;
  if (i < n * GAT_HC) out[i] = bias[i & (GAT_HC - 1)];
  if (i < e2)         ei_out[i] = ei_in[i];
  if (i < n * GAT_H)  { maxv[i] = -__builtin_inff(); denom[i] = 0.0f; }
}

// ---------------------------------------------------------------------------
// GEMM xp = x @ W via V_WMMA_F32_16X16X4_F32 (wave32).
// Block = 256 threads = 8 waves; W staged in LDS pre-swizzled into the WMMA
// B-fragment layout so each lane's (b.x,b.y) pair is one aligned ds_load_b64:
//   sW[k4*256 + col*4 + half*2 + lo] = W[(4*k4 + 2*half + lo)*64 + col]
// Per N-tile a wave's 32x8B reads hit dwords 4*l16+{0,1}/{2,3} -> 64 distinct
// banks, conflict-free.
// A frag (16x4): lanes 0-15 -> K = 4k+0,4k+1 ; lanes 16-31 -> K = 4k+2,4k+3.
// D (16x16 f32, 8 VGPRs): lane%16 = col, row = v + 8*(lane>=16).
// ---------------------------------------------------------------------------
__global__ __launch_bounds__(256) void gat_gemm(const float* __restrict__ x,
                                                const float* __restrict__ W,
                                                float* __restrict__ xp, int n) {
  __shared__ float sW[GAT_IN * GAT_HC];  // 32 KB, swizzled pair layout
  int t = threadIdx.x;
  for (int d = t; d < GAT_IN * GAT_HC; d += 256) {
    int k4   = d >> 8;
    int r    = d & 255;
    int col  = r >> 2;
    int half = (r >> 1) & 1;
    int lo   = d & 1;
    sW[d] = W[(4 * k4 + 2 * half + lo) * GAT_HC + col];
  }
  __syncthreads();

  int wave = t >> 5;
  int lane = t & 31;
  int half = lane >> 4;   // 0: lanes 0-15, 1: lanes 16-31
  int l16  = lane & 15;
  int row0 = (blockIdx.x * 8 + wave) * 16;

  int arow = row0 + l16;
  const float* xrow = x + (size_t)(arow < n ? arow : (n - 1)) * GAT_IN;
  const v2f* bp = (const v2f*)sW;

  v8f acc0 = {}, acc1 = {}, acc2 = {}, acc3 = {};
  for (int k4 = 0; k4 < GAT_IN / 4; ++k4) {
    v2f a = *(const v2f*)(xrow + 4 * k4 + 2 * half);  // 8B-aligned
    int base = k4 * 128 + l16 * 2 + half;             // v2f index
    v2f b0 = bp[base + 0];
    v2f b1 = bp[base + 32];
    v2f b2 = bp[base + 64];
    v2f b3 = bp[base + 96];
    acc0 = __builtin_amdgcn_wmma_f32_16x16x4_f32(false, a, false, b0, (short)0, acc0, false, false);
    acc1 = __builtin_amdgcn_wmma_f32_16x16x4_f32(false, a, false, b1, (short)0, acc1, false, false);
    acc2 = __builtin_amdgcn_wmma_f32_16x16x4_f32(false, a, false, b2, (short)0, acc2, false, false);
    acc3 = __builtin_amdgcn_wmma_f32_16x16x4_f32(false, a, false, b3, (short)0, acc3, false, false);
  }

  for (int v = 0; v < 8; ++v) {
    int row = row0 + v + 8 * half;
    if (row < n) {
      float* orow = xp + (size_t)row * GAT_HC + l16;
      orow[0]  = acc0[v];
      orow[16] = acc1[v];
      orow[32] = acc2[v];
      orow[48] = acc3[v];
    }
  }
}

// ---------------------------------------------------------------------------
// Per-node attention halves: a_src[n,h] = dot(xp[n,h,:], att_src[h,:]) etc.
// ---------------------------------------------------------------------------
__global__ void gat_node_att(const float* __restrict__ xp,
                             const float* __restrict__ att_src,
                             const float* __restrict__ att_dst,
                             float* __restrict__ a_src, float* __restrict__ a_dst,
                             int n) {
  int i = blockIdx.x * blockDim.x + threadIdx.x;
  if (i >= n * GAT_H) return;
  int h = i & (GAT_H - 1);
  const float* v = xp + (size_t)(i >> 2) * GAT_HC + h * GAT_C;
  float ss = 0.f, sd = 0.f;
  for (int c = 0; c < GAT_C; ++c) {
    float xv = v[c];
    ss += xv * att_src[h * GAT_C + c];
    sd += xv * att_dst[h * GAT_C + c];
  }
  a_src[i] = ss;
  a_dst[i] = sd;
}

// float atomic max via sign-split int/uint atomics (valid with -inf init)
__device__ __forceinline__ void atomicMaxF32(float* addr, float v) {
  if (v >= 0.0f) atomicMax((int*)addr, __float_as_int(v));
  else           atomicMin((unsigned int*)addr, __float_as_uint(v));
}

// ---------------------------------------------------------------------------
// Per-edge logits + leaky relu + segment max over dst.
// ---------------------------------------------------------------------------
__global__ void gat_edge_logits(const int* __restrict__ src, const int* __restrict__ dst,
                                const float* __restrict__ a_src,
                                const float* __restrict__ a_dst,
                                float* __restrict__ logits, float* __restrict__ maxv,
                                int E) {
  int i = blockIdx.x * blockDim.x + threadIdx.x;
  if (i >= E * GAT_H) return;
  int e = i >> 2, h = i & (GAT_H - 1);
  int s = src[e], d = dst[e];
  float l = a_src[s * GAT_H + h] + a_dst[d * GAT_H + h];
  l = (l > 0.0f) ? l : NEG_SLOPE * l;
  logits[i] = l;
  atomicMaxF32(maxv + d * GAT_H + h, l);
}

// Guard empty segments: -inf -> 0 (matches jnp.where(isfinite(m), m, 0))
__global__ void gat_guard(float* __restrict__ maxv, int n) {
  int i = blockIdx.x * blockDim.x + threadIdx.x;
  if (i >= n * GAT_H) return;
  float m = maxv[i];
  if (!(m > -__builtin_inff() && m < __builtin_inff())) maxv[i] = 0.0f;
}

// ---------------------------------------------------------------------------
// e = exp(logit - m[dst]); denom[dst] += e; store e in place of logits.
// ---------------------------------------------------------------------------
__global__ void gat_edge_exp(const int* __restrict__ dst,
                             float* __restrict__ ework,
                             const float* __restrict__ maxv,
                             float* __restrict__ denom, int E) {
  int i = blockIdx.x * blockDim.x + threadIdx.x;
  if (i >= E * GAT_H) return;
  int e = i >> 2, h = i & (GAT_H - 1);
  int d = dst[e];
  float ex = expf(ework[i] - maxv[d * GAT_H + h]);
  ework[i] = ex;
  atomicAdd(denom + d * GAT_H + h, ex);
}

// ---------------------------------------------------------------------------
// alpha = e / (denom[dst] + 1e-16); out[dst] += xp[src] * alpha.
// One thread per (edge, channel): 64 coalesced lanes per edge.
// ---------------------------------------------------------------------------
__global__ void gat_scatter(const int* __restrict__ src, const int* __restrict__ dst,
                            const float* __restrict__ xp,
                            const float* __restrict__ ework,
                            const float* __restrict__ denom,
                            float* __restrict__ out, float* __restrict__ alpha_out,
                            int E) {
  int i = blockIdx.x * blockDim.x + threadIdx.x;
  if (i >= E * GAT_HC) return;
  int e = i >> 6;
  int j = i & (GAT_HC - 1);
  int h = j >> 4;
  int s = src[e], d = dst[e];
  float al = ework[e * GAT_H + h] / (denom[d * GAT_H + h] + 1e-16f);
  if ((j & 15) == 0) alpha_out[e * GAT_H + h] = al;
  atomicAdd(out + (size_t)d * GAT_HC + j, xp[(size_t)s * GAT_HC + j] * al);
}

// ---------------------------------------------------------------------------
extern "C" void kernel_launch(void* const* d_in, const int* in_sizes, int n_in,
                              void* d_out, int out_size, void* d_ws, size_t ws_size,
                              hipStream_t stream) {
  const float* x       = (const float*)d_in[0];
  const float* W       = (const float*)d_in[1];
  const float* att_src = (const float*)d_in[2];
  const float* att_dst = (const float*)d_in[3];
  const float* bias    = (const float*)d_in[4];
  const int*   ei      = (const int*)d_in[5];

  int n = in_sizes[0] / GAT_IN;
  int E = in_sizes[5] / 2;
  const int* src = ei;
  const int* dst = ei + E;

  // tuple output layout: out [n*64] | edge_index [2E] | alpha [E*4]
  float* out_f     = (float*)d_out;
  int*   ei_out    = (int*)d_out + (size_t)n * GAT_HC;
  float* alpha_out = (float*)d_out + (size_t)n * GAT_HC + (size_t)2 * E;

  // workspace layout
  float* xp    = (float*)d_ws;                         // n*64
  float* a_src = xp    + (size_t)n * GAT_HC;           // n*4
  float* a_dst = a_src + (size_t)n * GAT_H;            // n*4
  float* maxv  = a_dst + (size_t)n * GAT_H;            // n*4
  float* denom = maxv  + (size_t)n * GAT_H;            // n*4
  float* ework = denom + (size_t)n * GAT_H;            // E*4 (logits -> exp)

  const int thr = 256;
  int initN = (n * GAT_HC > 2 * E) ? n * GAT_HC : 2 * E;
  gat_init<<<(initN + thr - 1) / thr, thr, 0, stream>>>(out_f, ei_out, ei, bias,
                                                        maxv, denom, n, 2 * E);

  int mblocks = (n + 127) / 128;  // 8 waves x 16 rows per block
  gat_gemm<<<mblocks, 256, 0, stream>>>(x, W, xp, n);

  gat_node_att<<<(n * GAT_H + thr - 1) / thr, thr, 0, stream>>>(xp, att_src, att_dst,
                                                                a_src, a_dst, n);

  gat_edge_logits<<<(E * GAT_H + thr - 1) / thr, thr, 0, stream>>>(src, dst, a_src,
                                                                   a_dst, ework, maxv, E);

  gat_guard<<<(n * GAT_H + thr - 1) / thr, thr, 0, stream>>>(maxv, n);

  gat_edge_exp<<<(E * GAT_H + thr - 1) / thr, thr, 0, stream>>>(dst, ework, maxv,
                                                                denom, E);

  gat_scatter<<<(E * GAT_HC + thr - 1) / thr, thr, 0, stream>>>(src, dst, xp, ework,
                                                                denom, out_f, alpha_out, E);
}